// E54DiagonalNoProj_47656957116853
// MI455X (gfx1250) — compile-verified
//
#include <hip/hip_runtime.h>
#include <cstddef>
#include <cstdint>

// ---------------------------------------------------------------------------
// E54 diagonal recurrence, chunked parallel scan for MI455X (gfx1250).
//   h_t = d * (silu(x_t) + h_{t-1}) + b      (d = sigmoid(log_d), per channel)
//   out_t = h_t * silu(h_t)
// 3 passes:  (1) per-chunk affine summaries  (2) tiny cross-chunk scan
//            (3) replay chunks and emit gated outputs.
// Passes 1/3 stage x through LDS via CDNA5 async global->LDS loads
// (ASYNCcnt + s_wait_asynccnt). Sigmoid uses gfx1250 v_tanh_f32 (1 trans op
// instead of exp+rcp). Output stream is stored non-temporally to keep L2 for x.
// ---------------------------------------------------------------------------

constexpr int L_CHUNK = 64;              // timesteps per chunk (T % 64 == 0)
constexpr int TPB     = 256;             // 8 wave32s per block
constexpr int GROUP   = 4;               // timesteps per async-load group
constexpr int NGROUPS = L_CHUNK / GROUP; // 16

#if defined(__AMDGCN__)
#  if __has_builtin(__builtin_amdgcn_global_load_async_to_lds_b128)
#    define HAVE_ASYNC_LDS 1
#  endif
#endif
#ifndef HAVE_ASYNC_LDS
#  define HAVE_ASYNC_LDS 0
#endif

typedef int   v4i __attribute__((ext_vector_type(4)));
typedef float v4f __attribute__((ext_vector_type(4)));

__device__ __forceinline__ float fsig(float z) {
#if defined(__AMDGCN__)
#  if __has_builtin(__builtin_amdgcn_tanhf)
  // sigmoid(z) = 0.5*tanh(z/2) + 0.5  -> single v_tanh_f32 (gfx1250)
  return fmaf(0.5f, __builtin_amdgcn_tanhf(0.5f * z), 0.5f);
#  else
  return __builtin_amdgcn_rcpf(1.0f + __expf(-z));  // v_exp_f32 + v_rcp_f32
#  endif
#else
  return 1.0f / (1.0f + __expf(-z));
#endif
}

__device__ __forceinline__ float4 sig4(float4 v) {
  return make_float4(fsig(v.x), fsig(v.y), fsig(v.z), fsig(v.w));
}

// h <- a * (silu(x) + h) + b
__device__ __forceinline__ void step4(float4& h, float4 xv, float4 a, float4 bb) {
  h.x = fmaf(a.x, xv.x * fsig(xv.x) + h.x, bb.x);
  h.y = fmaf(a.y, xv.y * fsig(xv.y) + h.y, bb.y);
  h.z = fmaf(a.z, xv.z * fsig(xv.z) + h.z, bb.z);
  h.w = fmaf(a.w, xv.w * fsig(xv.w) + h.w, bb.w);
}

// h * silu(h) = h*h*sigmoid(h)
__device__ __forceinline__ float4 selfgate4(float4 h) {
  return make_float4(h.x * h.x * fsig(h.x),
                     h.y * h.y * fsig(h.y),
                     h.z * h.z * fsig(h.z),
                     h.w * h.w * fsig(h.w));
}

// 16B non-temporal store (builtin requires a true vector type, not HIP float4)
__device__ __forceinline__ void nt_store4(float4 v, float* p) {
  v4f t;
  t.x = v.x; t.y = v.y; t.z = v.z; t.w = v.w;
  __builtin_nontemporal_store(t, (v4f*)p);
}

#if HAVE_ASYNC_LDS
__device__ __forceinline__ void async_cp_b128(const float4* g, float4* l) {
  __builtin_amdgcn_global_load_async_to_lds_b128(
      (v4i*)g, (v4i*)l, /*offset=*/0, /*cpol=*/0);
}
#endif

template <int N>
__device__ __forceinline__ void s_wait_asynccnt_le() {
#if defined(__AMDGCN__)
  asm volatile("s_wait_asynccnt %0" ::"n"(N) : "memory");
#endif
}

__device__ __forceinline__ void s_wait_dscnt0() {
#if defined(__AMDGCN__)
  asm volatile("s_wait_dscnt 0" ::: "memory");
#endif
}

// ---------------------------------------------------------------------------
// Pass 1 (WRITE_OUT=false): S_chunk via Horner (start from 0), store to ws.
// Pass 3 (WRITE_OUT=true):  start from h_in (read from ws), write gated out.
// Grid: (tiles, C, B).  One thread owns 4 consecutive channels of one chunk.
// ---------------------------------------------------------------------------
template <bool WRITE_OUT>
__global__ __launch_bounds__(TPB) void e54_chunk_kernel(
    const float* __restrict__ x, const float* __restrict__ log_d,
    const float* __restrict__ bias, float* __restrict__ ws,
    float* __restrict__ out, int T, int D, int C) {
  const int tid = threadIdx.x;
  const int c   = blockIdx.y;
  const int b   = blockIdx.z;
  const int ch  = (blockIdx.x * TPB + tid) * 4;
  if (ch >= D) return;
  const int t0 = c * L_CHUNK;

  const float4 a4 = sig4(*(const float4*)(log_d + ch));
  const float4 b4 = *(const float4*)(bias + ch);

  const size_t wsIdx = ((size_t)b * C + c) * (size_t)D + ch;
  float4 h;
  if (WRITE_OUT) h = *(const float4*)(ws + wsIdx);
  else           h = make_float4(0.f, 0.f, 0.f, 0.f);

  const float* xb = x + ((size_t)b * T + t0) * (size_t)D + ch;
  float* ob       = out + ((size_t)b * T + t0) * (size_t)D + ch;

#if HAVE_ASYNC_LDS
  // Double-buffered ring of GROUP timesteps; each lane owns its 16B slot, so
  // only the wave-local ASYNCcnt wait is needed (no barriers).
  __shared__ float4 ring[2][GROUP][TPB];
  auto issue = [&](int g) {
#pragma unroll
    for (int k = 0; k < GROUP; ++k) {
      const float4* src = (const float4*)(xb + (size_t)(g * GROUP + k) * D);
      async_cp_b128(src, &ring[g & 1][k][tid]);
    }
  };
  issue(0);
  issue(1);
  for (int g = 0; g < NGROUPS; ++g) {
    // Before consuming group g, only the (at most one) newer group may remain
    // in flight; on the last group drain fully.
    if (g + 1 < NGROUPS) s_wait_asynccnt_le<GROUP>();
    else                 s_wait_asynccnt_le<0>();
#pragma unroll
    for (int k = 0; k < GROUP; ++k) {
      float4 xv = ring[g & 1][k][tid];
      step4(h, xv, a4, b4);
      if (WRITE_OUT) {
        // write-once stream: non-temporal so it does not evict x from L2
        nt_store4(selfgate4(h), ob + (size_t)(g * GROUP + k) * D);
      }
    }
    if (g + 2 < NGROUPS) {
      s_wait_dscnt0();  // our ds reads of ring[g&1] must land before overwrite
      issue(g + 2);
    }
  }
#else
#pragma unroll 4
  for (int j = 0; j < L_CHUNK; ++j) {
    float4 xv = *(const float4*)(xb + (size_t)j * D);
    step4(h, xv, a4, b4);
    if (WRITE_OUT) nt_store4(selfgate4(h), ob + (size_t)j * D);
  }
#endif

  if (!WRITE_OUT) *(float4*)(ws + wsIdx) = h;  // chunk summary S
}

// ---------------------------------------------------------------------------
// Pass 2: per (b, 4-channel vec): h = h0; for each chunk c:
//   read S[c], overwrite with h (chunk-entry state), h = A*h + S,  A = d^L.
// Grid: (vpc/TPB, B). Writes h_final.
// ---------------------------------------------------------------------------
__global__ __launch_bounds__(TPB) void e54_prefix_kernel(
    const float* __restrict__ log_d, const float* __restrict__ h0,
    float* __restrict__ ws, float* __restrict__ h_final, int D, int C) {
  const int b  = blockIdx.y;
  const int ch = (blockIdx.x * TPB + threadIdx.x) * 4;
  if (ch >= D) return;

  const float4 a = sig4(*(const float4*)(log_d + ch));
  float4 A = make_float4(1.f, 1.f, 1.f, 1.f);
#pragma unroll
  for (int k = 0; k < L_CHUNK; ++k) {
    A.x *= a.x; A.y *= a.y; A.z *= a.z; A.w *= a.w;
  }

  float4 h = *(const float4*)(h0 + (size_t)b * D + ch);
  for (int c = 0; c < C; ++c) {
    const size_t w = ((size_t)b * C + c) * (size_t)D + ch;
    float4 S = *(const float4*)(ws + w);
    *(float4*)(ws + w) = h;  // becomes h_in for chunk c (in-place swap)
    h.x = fmaf(A.x, h.x, S.x);
    h.y = fmaf(A.y, h.y, S.y);
    h.z = fmaf(A.z, h.z, S.z);
    h.w = fmaf(A.w, h.w, S.w);
  }
  *(float4*)(h_final + (size_t)b * D + ch) = h;
}

// ---------------------------------------------------------------------------
extern "C" void kernel_launch(void* const* d_in, const int* in_sizes, int n_in,
                              void* d_out, int out_size, void* d_ws,
                              size_t ws_size, hipStream_t stream) {
  const float* x     = (const float*)d_in[0];
  const float* h0    = (const float*)d_in[1];
  const float* log_d = (const float*)d_in[2];
  const float* bias  = (const float*)d_in[3];

  const int D  = in_sizes[2];
  const int BD = in_sizes[1];
  const int B  = BD / D;
  const int T  = in_sizes[0] / BD;
  const int C  = T / L_CHUNK;  // T=4096 -> 64 chunks of 64
  const int vpc   = D / 4;
  const int tiles = (vpc + TPB - 1) / TPB;

  float* out  = (float*)d_out;
  float* hfin = out + (size_t)B * T * D;
  float* ws   = (float*)d_ws;  // B*C*D floats (4 MB for this shape)

  dim3 blk(TPB);

  // Pass 1: chunk summaries S -> ws
  e54_chunk_kernel<false><<<dim3(tiles, C, B), blk, 0, stream>>>(
      x, log_d, bias, ws, /*out unused*/ ws, T, D, C);

  // Pass 2: cross-chunk scan; ws becomes per-chunk entry states; h_final out
  e54_prefix_kernel<<<dim3(tiles, B), blk, 0, stream>>>(
      log_d, h0, ws, hfin, D, C);

  // Pass 3: replay chunks from entry states, write gated outputs
  e54_chunk_kernel<true><<<dim3(tiles, C, B), blk, 0, stream>>>(
      x, log_d, bias, ws, out, T, D, C);
}